// GCN_80358838108317
// MI455X (gfx1250) — compile-verified
//
#include <hip/hip_runtime.h>
#include <hip/hip_bf16.h>

// ---------------------------------------------------------------------------
// GCN 2-layer forward for MI455X (gfx1250, wave32).
//  * GEMMs via V_WMMA_F32_16X16X4_F32 (full fp32 precision; GEMM is ~1% of
//    the roofline cost here, so no reason to quantize).
//  * Edge aggregation via inline-asm GLOBAL_ATOMIC_ADD_F32 (no-return form,
//    STOREcnt-tracked) -- guarantees the native L2 FP atomic, never a CAS loop.
//    Working sets (h: 51MB, out: 51MB) are L2-resident (192MB), so the ~1.7GB
//    of gather+atomic traffic per layer stays on-chip at L2 bandwidth.
//  * Self-loops + bias folded into a per-node init pass (norm_self = dis[n]^2).
// ---------------------------------------------------------------------------

#define NNODES 50000
#define NEDGES 800000
#define NFEAT  128
#define NHID   256
#define NOUT   128

typedef float v2f __attribute__((ext_vector_type(2)));
typedef float v8f __attribute__((ext_vector_type(8)));

// Fire-and-forget f32 atomic add: emits GLOBAL_ATOMIC_ADD_F32 (no return ->
// tracked on STOREcnt; s_endpgm's implicit wait-idle covers completion).
__device__ __forceinline__ void atomic_fadd(float* p, float v) {
    asm volatile("global_atomic_add_f32 %0, %1, off" :: "v"(p), "v"(v) : "memory");
}

// ---- degree / normalization --------------------------------------------------
__global__ void k_deg_init(float* __restrict__ deg, int n) {
    int i = blockIdx.x * blockDim.x + threadIdx.x;
    if (i < n) deg[i] = 1.0f;                      // self-loop weight
}

__global__ void k_deg_acc(const int* __restrict__ dst, const float* __restrict__ w,
                          float* __restrict__ deg, int nE) {
    int e = blockIdx.x * blockDim.x + threadIdx.x;
    if (e < nE) atomic_fadd(&deg[dst[e]], w[e]);
}

__global__ void k_dis(const float* __restrict__ deg, float* __restrict__ dis, int n) {
    int i = blockIdx.x * blockDim.x + threadIdx.x;
    if (i < n) {
        float d = deg[i];
        dis[i] = (d > 0.0f) ? rsqrtf(d) : 0.0f;
    }
}

// ---- H = X @ W^T via fp32 WMMA ----------------------------------------------
// One wave computes a 16-row strip of H across the full output width (NO cols),
// reusing each A fragment across NO/16 column tiles.
template<int K, int NO, bool SANITIZE>
__global__ __launch_bounds__(256) void k_gemm_xWT(const float* __restrict__ X,
                                                  const float* __restrict__ W,
                                                  float* __restrict__ H,
                                                  int nTiles) {
    const int lane = threadIdx.x & 31;
    const int wave = threadIdx.x >> 5;
    const int tile = blockIdx.x * 8 + wave;
    if (tile >= nTiles) return;                    // wave-uniform: EXEC stays all-1s

    const int l15  = lane & 15;
    const int half = lane >> 4;                    // 0 -> K=0,1 ; 1 -> K=2,3
    const int rowBase = tile * 16;

    constexpr int NT = NO / 16;
    v8f acc[NT] = {};

    const float* xrow = X + (size_t)(rowBase + l15) * K;

    #pragma unroll 1
    for (int kk = 0; kk < K; kk += 4) {
        // A fragment: 32-bit 16x4 layout (ISA 7.12.2), one 8B-aligned float2 load
        v2f a = *reinterpret_cast<const v2f*>(xrow + kk + 2 * half);
        if (SANITIZE) {                            // nan_to_num(x) from reference
            a.x = (a.x == a.x) ? a.x : 0.0f;
            a.y = (a.y == a.y) ? a.y : 0.0f;
        }
        #pragma unroll
        for (int t = 0; t < NT; ++t) {
            const int col = t * 16 + l15;          // B[k][n] = W[n*K + k]  (W^T)
            v2f b = *reinterpret_cast<const v2f*>(W + (size_t)col * K + kk + 2 * half);
            acc[t] = __builtin_amdgcn_wmma_f32_16x16x4_f32(
                false, a, false, b, (short)0, acc[t], false, false);
        }
    }

    // C/D layout: VGPR r holds row r (lanes 0-15) / row r+8 (lanes 16-31)
    #pragma unroll
    for (int t = 0; t < NT; ++t) {
        const int col = t * 16 + l15;
        #pragma unroll
        for (int r = 0; r < 8; ++r) {
            H[(size_t)(rowBase + r + 8 * half) * NO + col] = acc[t][r];
        }
    }
}

// ---- out[i] = bias + h[i] * dis[i]^2   (self-loop + bias fold) ---------------
template<int F>
__global__ void k_init_out(const float* __restrict__ H, const float* __restrict__ dis,
                           const float* __restrict__ bias, float* __restrict__ out,
                           int n) {
    int idx = blockIdx.x * blockDim.x + threadIdx.x;
    if (idx >= n * F) return;
    int i = idx / F, j = idx % F;
    float s = dis[i];
    out[idx] = bias[j] + H[idx] * s * s;
}

// ---- edge scatter: out[dst] += h[src] * (dis[src]*w*dis[dst]) ----------------
// One wave per edge; lane l handles float4 chunk l of each 128-float block.
template<int F>
__global__ __launch_bounds__(256) void k_edge_agg(const float* __restrict__ H,
                                                  const float* __restrict__ dis,
                                                  const int* __restrict__ src,
                                                  const int* __restrict__ dst,
                                                  const float* __restrict__ w,
                                                  float* __restrict__ out, int nE) {
    const int lane = threadIdx.x & 31;
    const int e = (int)((blockIdx.x * blockDim.x + threadIdx.x) >> 5);
    if (e >= nE) return;

    const int s = src[e];
    const int d = dst[e];
    const float nrm = dis[s] * w[e] * dis[d];

    const float4* hs = reinterpret_cast<const float4*>(H + (size_t)s * F);
    float* od = out + (size_t)d * F;

    #pragma unroll
    for (int c = 0; c < F / 128; ++c) {
        float4 v = hs[c * 32 + lane];              // coalesced 512B/wave gather
        int base = (c * 32 + lane) * 4;
        atomic_fadd(od + base + 0, v.x * nrm);
        atomic_fadd(od + base + 1, v.y * nrm);
        atomic_fadd(od + base + 2, v.z * nrm);
        atomic_fadd(od + base + 3, v.w * nrm);
    }
}

// ---- mean + max pool over nodes, one block per feature -----------------------
__global__ void k_pool(const float* __restrict__ H, float* __restrict__ out,
                       int n, int F) {
    const int j = blockIdx.x;
    float s = 0.0f, m = -3.402823466e38f;
    for (int i = threadIdx.x; i < n; i += blockDim.x) {
        float v = H[(size_t)i * F + j];
        s += v;
        m = fmaxf(m, v);
    }
    __shared__ float ss[256];
    __shared__ float sm[256];
    ss[threadIdx.x] = s;
    sm[threadIdx.x] = m;
    __syncthreads();
    for (int off = 128; off > 0; off >>= 1) {
        if (threadIdx.x < off) {
            ss[threadIdx.x] += ss[threadIdx.x + off];
            sm[threadIdx.x] = fmaxf(sm[threadIdx.x], sm[threadIdx.x + off]);
        }
        __syncthreads();
    }
    if (threadIdx.x == 0) {
        out[j]     = ss[0] / (float)n;             // pooled_avg
        out[F + j] = sm[0];                        // pooled_max
    }
}

// ---------------------------------------------------------------------------
extern "C" void kernel_launch(void* const* d_in, const int* in_sizes, int n_in,
                              void* d_out, int out_size, void* d_ws, size_t ws_size,
                              hipStream_t stream) {
    (void)in_sizes; (void)n_in; (void)out_size; (void)ws_size;

    const float* x  = (const float*)d_in[0];       // [50000,128]
    const int*   ei = (const int*)d_in[1];         // [2,800000] (int32 after JAX x32)
    const float* ew = (const float*)d_in[2];       // [800000]
    const float* W1 = (const float*)d_in[3];       // [256,128]
    const float* b1 = (const float*)d_in[4];       // [256]
    const float* W2 = (const float*)d_in[5];       // [128,256]
    const float* b2 = (const float*)d_in[6];       // [128]
    const int* srcE = ei;
    const int* dstE = ei + NEDGES;

    // workspace layout (~103 MB): deg | dis | h1 (reused as h2) | o1 (reused as o2)
    float* deg = (float*)d_ws;
    float* dis = deg + 50048;
    float* h1  = dis + 50048;
    float* o1  = h1 + (size_t)NNODES * NHID;
    float* h2  = h1;   // h1 dead after layer-1 aggregation
    float* o2  = o1;   // o1 dead after GEMM2

    const int T = 256;
    const int nTiles = NNODES / 16;                // 3125 (exact)

    k_deg_init<<<(NNODES + T - 1) / T, T, 0, stream>>>(deg, NNODES);
    k_deg_acc<<<(NEDGES + T - 1) / T, T, 0, stream>>>(dstE, ew, deg, NEDGES);
    k_dis<<<(NNODES + T - 1) / T, T, 0, stream>>>(deg, dis, NNODES);

    // layer 1: h1 = nan_to_num(x) @ W1^T ; o1 = segsum + self-loop + b1
    k_gemm_xWT<NFEAT, NHID, true><<<(nTiles + 7) / 8, 256, 0, stream>>>(x, W1, h1, nTiles);
    k_init_out<NHID><<<(NNODES * NHID + T - 1) / T, T, 0, stream>>>(h1, dis, b1, o1, NNODES);
    k_edge_agg<NHID><<<(NEDGES + 7) / 8, 256, 0, stream>>>(h1, dis, srcE, dstE, ew, o1, NEDGES);

    // layer 2: h2 = o1 @ W2^T ; o2 = segsum + self-loop + b2
    k_gemm_xWT<NHID, NOUT, false><<<(nTiles + 7) / 8, 256, 0, stream>>>(o1, W2, h2, nTiles);
    k_init_out<NOUT><<<(NNODES * NOUT + T - 1) / T, T, 0, stream>>>(h2, dis, b2, o2, NNODES);
    k_edge_agg<NOUT><<<(NEDGES + 7) / 8, 256, 0, stream>>>(h2, dis, srcE, dstE, ew, o2, NEDGES);

    // pooled_avg(128) ++ pooled_max(128)
    k_pool<<<NOUT, 256, 0, stream>>>(o2, (float*)d_out, NNODES, NOUT);
}